// GCNModel_3951369912906
// MI455X (gfx1250) — compile-verified
//
#include <hip/hip_runtime.h>
#include <hip/hip_bf16.h>

#define NN   100000
#define NE   1600000
#define NG   64
#define INC  9
#define HID  128
#define OUTC 2
#define NMSG (NE + NN)
#define LDA  132   // padded LDS stride for A tile (128 + 4) to avoid bank conflicts

typedef __attribute__((ext_vector_type(2))) float v2f;
typedef __attribute__((ext_vector_type(8))) float v8f;

// ---------------- degree / normalization ----------------
__global__ void deg_kernel(const long long* __restrict__ col, float* __restrict__ deg) {
    int e = blockIdx.x * blockDim.x + threadIdx.x;
    if (e < NE) atomicAdd(&deg[(int)col[e]], 1.0f);
}

__global__ void dinv_kernel(const float* __restrict__ deg, float* __restrict__ dinv) {
    int i = blockIdx.x * blockDim.x + threadIdx.x;
    if (i < NN) dinv[i] = rsqrtf(deg[i] + 1.0f);   // +1 accounts for the self-loop
}

// ---------------- layer-1 GEMM (K=9, VALU; write-bound) ----------------
__global__ void __launch_bounds__(HID) gemm1_kernel(const float* __restrict__ x,
                                                    const float* __restrict__ W1,
                                                    float* __restrict__ out) {
    int i = blockIdx.x;        // node
    int j = threadIdx.x;       // feature
    float acc = 0.f;
#pragma unroll
    for (int k = 0; k < INC; ++k)
        acc += x[i * INC + k] * W1[k * HID + j];
    out[(size_t)i * HID + j] = acc;
}

// ---------------- edge aggregation: wave32 per message ----------------
__global__ void __launch_bounds__(256) aggregate_kernel(const float* __restrict__ src,
                                                        const long long* __restrict__ rows,
                                                        const long long* __restrict__ cols,
                                                        const float* __restrict__ dinv,
                                                        float* __restrict__ dst) {
    int m    = blockIdx.x * 8 + (threadIdx.x >> 5);  // 8 waves / block
    int lane = threadIdx.x & 31;
    if (m >= NMSG) return;
    int r, c;
    if (m < NE) { r = (int)rows[m]; c = (int)cols[m]; }
    else        { r = c = m - NE; }                  // self-loop
    float norm = dinv[r] * dinv[c];
    const float4 v = ((const float4*)(src + (size_t)r * HID))[lane]; // coalesced 512B row
    float* d = dst + (size_t)c * HID + lane * 4;
    atomicAdd(d + 0, v.x * norm);
    atomicAdd(d + 1, v.y * norm);
    atomicAdd(d + 2, v.z * norm);
    atomicAdd(d + 3, v.w * norm);
}

__global__ void bias_relu_kernel(float* __restrict__ h, const float* __restrict__ b) {
    int idx = blockIdx.x * blockDim.x + threadIdx.x;
    if (idx < NN * HID) {
        float v = h[idx] + b[idx & (HID - 1)];
        h[idx] = v > 0.f ? v : 0.f;
    }
}

// ---------------- layer-2 GEMM: f32 WMMA 16x16x4 ----------------
// Block: 256 threads = 8 waves; block owns a 16-row strip, wave w owns cols [16w,16w+16).
__global__ void __launch_bounds__(256) gemm2_wmma_kernel(const float* __restrict__ A,  // [NN,HID]
                                                         const float* __restrict__ W,  // [HID,HID]
                                                         float* __restrict__ out) {    // [NN,HID]
    __shared__ float lA[16 * LDA];     //  8.25 KB, padded stride
    __shared__ float lB[HID * HID];    // 64 KB (all of W2)
    const int tid  = threadIdx.x;
    const int row0 = blockIdx.x * 16;

    // stage W2 -> LDS (linear copy, float4)
    {
        const float4* W4  = (const float4*)W;
        float4*       lB4 = (float4*)lB;
        for (int t = tid; t < (HID * HID) / 4; t += 256) lB4[t] = W4[t];
    }
    // stage 16x128 A tile -> LDS with padded stride
    {
        const float4* A4 = (const float4*)(A + (size_t)row0 * HID);
        for (int t = tid; t < (16 * HID) / 4; t += 256) {
            int r  = t >> 5;    // 32 float4 per row
            int cc = t & 31;
            ((float4*)(lA + r * LDA))[cc] = A4[t];
        }
    }
    __syncthreads();

    const int wave = tid >> 5;
    const int lane = tid & 31;
    const int half = lane >> 4;   // 0: K pair {k,k+1} / rows v ; 1: {k+2,k+3} / rows v+8
    const int l16  = lane & 15;
    const int col0 = wave * 16;

    v8f c = {};
#pragma unroll 4
    for (int k = 0; k < HID; k += 4) {
        const int kk = k + half * 2;
        v2f a, b;
        // A frag (16x4 f32): lane row = l16, K = kk, kk+1
        a.x = lA[l16 * LDA + kk];
        a.y = lA[l16 * LDA + kk + 1];
        // B frag (4x16 f32): row K = kk / kk+1, col = col0 + l16 (conflict-free)
        b.x = lB[kk * HID + col0 + l16];
        b.y = lB[(kk + 1) * HID + col0 + l16];
        c = __builtin_amdgcn_wmma_f32_16x16x4_f32(false, a, false, b,
                                                  (short)0, c, false, false);
    }

    // C/D layout: VGPR v -> row v (lanes 0-15) or v+8 (lanes 16-31), col = l16
    float* o = out + (size_t)row0 * HID + col0 + l16;
#pragma unroll
    for (int v = 0; v < 8; ++v) {
        int m = v + half * 8;
        o[(size_t)m * HID] = c[v];
    }
}

// ---------------- pooling + classifier ----------------
__global__ void pool_sum_kernel(const float* __restrict__ h,
                                const long long* __restrict__ batch,
                                float* __restrict__ sums) {
    int idx = blockIdx.x * blockDim.x + threadIdx.x;
    if (idx < NN * HID) {
        int i = idx >> 7;
        int j = idx & (HID - 1);
        atomicAdd(&sums[(int)batch[i] * HID + j], h[idx]);
    }
}

__global__ void pool_cnt_kernel(const long long* __restrict__ batch, float* __restrict__ cnt) {
    int i = blockIdx.x * blockDim.x + threadIdx.x;
    if (i < NN) atomicAdd(&cnt[(int)batch[i]], 1.0f);
}

__global__ void classify_kernel(const float* __restrict__ sums, const float* __restrict__ cnt,
                                const float* __restrict__ Wc, const float* __restrict__ bc,
                                float* __restrict__ out) {
    int t = threadIdx.x;            // 128 threads = 64 graphs x 2 outputs
    int g = t >> 1, o = t & 1;
    float inv = 1.0f / fmaxf(cnt[g], 1.0f);
    float acc = bc[o];
    for (int k = 0; k < HID; ++k)
        acc += sums[g * HID + k] * inv * Wc[k * OUTC + o];
    out[g * OUTC + o] = acc;
}

// ---------------- launcher ----------------
extern "C" void kernel_launch(void* const* d_in, const int* in_sizes, int n_in,
                              void* d_out, int out_size, void* d_ws, size_t ws_size,
                              hipStream_t stream) {
    const float*     x     = (const float*)d_in[0];
    const long long* eidx  = (const long long*)d_in[1];  // [2, NE] int64
    const long long* batch = (const long long*)d_in[2];
    const float*     W1    = (const float*)d_in[3];
    const float*     b1    = (const float*)d_in[4];
    const float*     W2    = (const float*)d_in[5];
    const float*     b2    = (const float*)d_in[6];
    const float*     Wc    = (const float*)d_in[7];
    const float*     bc    = (const float*)d_in[8];
    float*           out   = (float*)d_out;

    const long long* rows = eidx;        // edge_index[0] (sources)
    const long long* cols = eidx + NE;   // edge_index[1] (targets)

    // workspace layout (all float, offsets keep 16B alignment)
    float* ws   = (float*)d_ws;
    float* deg  = ws;                    ws += NN;
    float* dinv = ws;                    ws += NN;
    float* sums = ws;                    ws += NG * HID;
    float* cnt  = ws;                    ws += NG;
    float* bufA = ws;                    ws += (size_t)NN * HID;
    float* bufB = ws;

    const size_t featB = (size_t)NN * HID * sizeof(float);

    // degrees + symmetric normalization
    hipMemsetAsync(deg, 0, NN * sizeof(float), stream);
    deg_kernel <<<(NE + 255) / 256, 256, 0, stream>>>(cols, deg);
    dinv_kernel<<<(NN + 255) / 256, 256, 0, stream>>>(deg, dinv);

    // layer 1: x@W1 -> aggregate -> +b1, relu
    gemm1_kernel<<<NN, HID, 0, stream>>>(x, W1, bufA);
    hipMemsetAsync(bufB, 0, featB, stream);
    aggregate_kernel<<<(NMSG + 7) / 8, 256, 0, stream>>>(bufA, rows, cols, dinv, bufB);
    bias_relu_kernel<<<(NN * HID + 255) / 256, 256, 0, stream>>>(bufB, b1);

    // layer 2: h@W2 (WMMA f32) -> aggregate -> +b2, relu
    gemm2_wmma_kernel<<<NN / 16, 256, 0, stream>>>(bufB, W2, bufA);
    hipMemsetAsync(bufB, 0, featB, stream);
    aggregate_kernel<<<(NMSG + 7) / 8, 256, 0, stream>>>(bufA, rows, cols, dinv, bufB);
    bias_relu_kernel<<<(NN * HID + 255) / 256, 256, 0, stream>>>(bufB, b2);

    // mean pool + classifier
    hipMemsetAsync(sums, 0, (NG * HID + NG) * sizeof(float), stream);  // sums + cnt contiguous
    pool_sum_kernel<<<(NN * HID + 255) / 256, 256, 0, stream>>>(bufB, batch, sums);
    pool_cnt_kernel<<<(NN + 255) / 256, 256, 0, stream>>>(batch, cnt);
    classify_kernel<<<1, 128, 0, stream>>>(sums, cnt, Wc, bc, out);
}